// SNN_CNN_Hybrid_489626272337
// MI455X (gfx1250) — compile-verified
//
#include <hip/hip_runtime.h>
#include <hip/hip_bf16.h>
#include <math.h>

#define DEV static __device__ __forceinline__

typedef float v2f __attribute__((ext_vector_type(2)));
typedef float v8f __attribute__((ext_vector_type(8)));

// ---------------- problem constants ----------------
constexpr int T_ = 200, B_ = 2;
constexpr int H1 = 100, W1 = 367;   // layer1 resolution
constexpr int H2 = 50,  W2 = 183;   // layer2 (= pooled layer1)
constexpr int H3 = 25,  W3 = 91;    // layer3
constexpr int H4 = 12,  W4 = 45;    // layer4
constexpr int C1_ = 8, C2_ = 16, C3_ = 32, C4_ = 32;
constexpr int CC  = C1_ + C2_ + C3_ + C4_;   // 88
constexpr float BETA = 0.9f, THR = 1.0f;

// ---------------- workspace layout (bytes) ----------------
// Pooled spike counts, 4 bits/channel, packed 8 channels per u32.
constexpr size_t P1_WORDS = (size_t)T_ * B_ * H2 * W2;       //  8 ch -> 1 word
constexpr size_t P2_WORDS = (size_t)T_ * B_ * H3 * W3 * 2;   // 16 ch -> 2 words
constexpr size_t P3_WORDS = (size_t)T_ * B_ * H4 * W4 * 4;   // 32 ch -> 4 words
constexpr size_t S1_N = (size_t)B_ * C1_ * H1 * W1;
constexpr size_t S2_N = (size_t)B_ * C2_ * H2 * W2;
constexpr size_t S3_N = (size_t)B_ * C3_ * H3 * W3;
constexpr size_t S4_N = (size_t)B_ * C4_ * H4 * W4;
constexpr size_t oP1 = 0;
constexpr size_t oP2 = oP1 + P1_WORDS * 4;
constexpr size_t oP3 = oP2 + P2_WORDS * 4;
constexpr size_t oS1 = oP3 + P3_WORDS * 4;
constexpr size_t oS2 = oS1 + S1_N * 4;
constexpr size_t oS3 = oS2 + S2_N * 4;
constexpr size_t oS4 = oS3 + S3_N * 4;

// =====================================================================
// Layer 1: conv(1->8) + LIF over T (scalar; 72 MAC/t, x-load latency bound)
// Block = 128 threads = 16(w) x 8(h) pixel tile; pooling via LDS.
// =====================================================================
__global__ __launch_bounds__(128) void k_layer1(const float* __restrict__ x,
    const float* __restrict__ w1, const float* __restrict__ b1,
    unsigned* __restrict__ P1, float* __restrict__ S1) {
  __shared__ float wLDS[C1_ * 9];
  __shared__ float bLDS[C1_];
  __shared__ unsigned maskLDS[128];
  const int tid = threadIdx.x;
  for (int i = tid; i < C1_ * 9; i += 128) wLDS[i] = w1[i];
  if (tid < C1_) bLDS[tid] = b1[tid];
  __syncthreads();

  const int tw = tid & 15, th = tid >> 4;
  const int w = blockIdx.x * 16 + tw;
  const int h = blockIdx.y * 8 + th;
  const int b = blockIdx.z;
  const bool act  = (w < W1) && (h < H1);
  const bool pool = act && !(tw & 1) && !(th & 1) && ((w >> 1) < W2) && ((h >> 1) < H2);
  const int pw = w >> 1, ph = h >> 1;

  float mem[C1_], sum[C1_];
#pragma unroll
  for (int c = 0; c < C1_; ++c) { mem[c] = 0.f; sum[c] = 0.f; }

  for (int t = 0; t < T_; ++t) {
    const float* xt = x + (size_t)(t * B_ + b) * (H1 * W1);
    float xv[9];
#pragma unroll
    for (int dy = -1; dy <= 1; ++dy)
#pragma unroll
      for (int dx = -1; dx <= 1; ++dx) {
        const int hh = h + dy, ww = w + dx;
        float v = 0.f;
        if (act && hh >= 0 && hh < H1 && ww >= 0 && ww < W1) v = xt[hh * W1 + ww];
        xv[(dy + 1) * 3 + (dx + 1)] = v;
      }
    if (t + 1 < T_)  // stream next timestep through L2 (global_prefetch_b8)
      __builtin_prefetch(x + (size_t)((t + 1) * B_ + b) * (H1 * W1) +
                             (act ? h * W1 + w : 0), 0, 1);

    unsigned m = 0u;
#pragma unroll
    for (int c = 0; c < C1_; ++c) {
      float cur = bLDS[c];
#pragma unroll
      for (int k = 0; k < 9; ++k) cur = fmaf(wLDS[c * 9 + k], xv[k], cur);
      const float mm = BETA * mem[c] + cur;
      const float sp = (mm >= THR) ? 1.f : 0.f;
      sum[c] += sp;
      mem[c] = mm - sp * THR;
      m |= ((mm >= THR) ? 1u : 0u) << c;
    }
    maskLDS[tid] = m;
    __syncthreads();
    if (pool) {
      const unsigned m00 = maskLDS[tid],      m01 = maskLDS[tid + 1];
      const unsigned m10 = maskLDS[tid + 16], m11 = maskLDS[tid + 17];
      unsigned word = 0u;
#pragma unroll
      for (int c = 0; c < C1_; ++c) {
        const unsigned cnt = ((m00 >> c) & 1u) + ((m01 >> c) & 1u) +
                             ((m10 >> c) & 1u) + ((m11 >> c) & 1u);
        word |= cnt << (4 * c);
      }
      P1[((size_t)(t * B_ + b) * H2 + ph) * W2 + pw] = word;
    }
    __syncthreads();
  }
  if (act) {
#pragma unroll
    for (int c = 0; c < C1_; ++c)
      S1[(size_t)(b * C1_ + c) * (H1 * W1) + h * W1 + w] = sum[c];
  }
}

// =====================================================================
// Layer 2 (WMMA): conv(8->16) + LIF. K = 9*8 = 72 -> 18 WMMA steps.
// Wave = 16 pixels (one row) x 16 out channels; C/D layout holds the LIF
// state (lane = channel, row = pixel). Block = 2 waves = 2 rows (for 2x2
// pooling). A fragment = nibble extract from 9 cached tap words; B
// fragment pre-permuted into LDS, one ds_load_b64 per step.
// =====================================================================
__global__ __launch_bounds__(64) void k_layer2(const unsigned* __restrict__ P1,
    const float* __restrict__ w2, const float* __restrict__ b2,
    unsigned* __restrict__ P2, float* __restrict__ S2) {
  constexpr int NK = 72 / 4;  // 18
  __shared__ float Bf[NK * 32 * 2];
  __shared__ float bLDS[C2_];
  __shared__ unsigned maskLDS[2][16];
  const int tid = threadIdx.x;
  for (int ent = tid; ent < NK * 32; ent += 64) {
    const int el = ent & 31, kk = ent >> 5;
    const int n = el & 15;
    const int k0 = kk * 4 + ((el >> 4) << 1);
#pragma unroll
    for (int j = 0; j < 2; ++j) {
      const int k = k0 + j;                 // k = tap*C1 + ci
      const int tap = k / C1_, ci = k % C1_;
      Bf[ent * 2 + j] = w2[(n * C1_ + ci) * 9 + tap];
    }
  }
  if (tid < C2_) bLDS[tid] = b2[tid];
  __syncthreads();

  const int lane = tid & 31;
  const int wrow = tid >> 5;                 // row within 2-row block
  const int n = lane & 15;
  const int shiftbase = (lane >> 4) << 3;    // 4*khalf
  const int wpix = blockIdx.x * 16 + n;      // A-fragment pixel column
  const int h = blockIdx.y * 2 + wrow;       // always < H2 (grid.y = 25)
  const int b = blockIdx.z;
  const float bias = bLDS[n];

  v8f mem = {0.f,0.f,0.f,0.f,0.f,0.f,0.f,0.f};
  v8f sum = {0.f,0.f,0.f,0.f,0.f,0.f,0.f,0.f};

  for (int t = 0; t < T_; ++t) {
    unsigned wd[9];
#pragma unroll
    for (int dy = -1; dy <= 1; ++dy)
#pragma unroll
      for (int dx = -1; dx <= 1; ++dx) {
        const int hh = h + dy, ww = wpix + dx;
        unsigned v = 0u;
        if (hh >= 0 && hh < H2 && ww >= 0 && ww < W2)
          v = P1[((size_t)(t * B_ + b) * H2 + hh) * W2 + ww];
        wd[(dy + 1) * 3 + (dx + 1)] = v;
      }
    v8f acc = {bias,bias,bias,bias,bias,bias,bias,bias};
#pragma unroll
    for (int kk = 0; kk < NK; ++kk) {
      const unsigned wsel = wd[kk >> 1];
      const int s = ((kk & 1) << 4) + shiftbase;
      v2f a;
      a.x = (float)((wsel >> s) & 0xFu) * 0.25f;
      a.y = (float)((wsel >> (s + 4)) & 0xFu) * 0.25f;
      const v2f bb = *reinterpret_cast<const v2f*>(&Bf[(kk * 32 + lane) * 2]);
      acc = __builtin_amdgcn_wmma_f32_16x16x4_f32(false, a, false, bb,
                                                  (short)0, acc, false, false);
    }
#pragma unroll
    for (int r = 0; r < 8; ++r) {
      const float mm = BETA * mem[r] + acc[r];
      const bool sp = (mm >= THR);
      sum[r] += sp ? 1.f : 0.f;
      mem[r] = mm - (sp ? THR : 0.f);
      const unsigned bal = (unsigned)__ballot(sp);
      if (lane == 0) {            // bits 0-15: pixel r; bits 16-31: pixel 8+r
        maskLDS[wrow][r] = bal & 0xFFFFu;
        maskLDS[wrow][8 + r] = bal >> 16;
      }
    }
    __syncthreads();
    if (tid < 8) {
      const int pw = blockIdx.x * 8 + tid;
      if (pw < W3) {
        const int px = 2 * tid;
        const unsigned m00 = maskLDS[0][px], m01 = maskLDS[0][px + 1];
        const unsigned m10 = maskLDS[1][px], m11 = maskLDS[1][px + 1];
        unsigned wordv[2] = {0u, 0u};
#pragma unroll
        for (int c = 0; c < C2_; ++c) {
          const unsigned cnt = ((m00 >> c) & 1u) + ((m01 >> c) & 1u) +
                               ((m10 >> c) & 1u) + ((m11 >> c) & 1u);
          wordv[c >> 3] |= cnt << (4 * (c & 7));
        }
        const size_t base =
            (((size_t)(t * B_ + b) * H3 + blockIdx.y) * W3 + pw) * 2;
        P2[base + 0] = wordv[0];
        P2[base + 1] = wordv[1];
      }
    }
    __syncthreads();
  }
#pragma unroll
  for (int r = 0; r < 8; ++r) {
    const int m = r + ((lane >> 4) << 3);
    const int w = blockIdx.x * 16 + m;
    if (w < W2) S2[(size_t)(b * C2_ + n) * (H2 * W2) + h * W2 + w] = sum[r];
  }
}

// =====================================================================
// Layer 3 (WMMA): conv(16->32) + LIF. K = 144 -> 36 steps.
// Block = 4 waves = 2 rows x 2 out-channel tiles.
// =====================================================================
__global__ __launch_bounds__(128) void k_layer3(const unsigned* __restrict__ P2,
    const float* __restrict__ w3, const float* __restrict__ b3,
    unsigned* __restrict__ P3, float* __restrict__ S3) {
  constexpr int NK = 144 / 4;  // 36
  __shared__ float Bf[2 * NK * 32 * 2];   // per co-tile fragments
  __shared__ float bLDS[C3_];
  __shared__ unsigned maskLDS[2][2][16];  // [row][cotile][pixel]
  const int tid = threadIdx.x;
  for (int ent = tid; ent < 2 * NK * 32; ent += 128) {
    const int el = ent & 31;
    const int kk = (ent >> 5) % NK;
    const int ct = (ent >> 5) / NK;
    const int n = el & 15;
    const int k0 = kk * 4 + ((el >> 4) << 1);
#pragma unroll
    for (int j = 0; j < 2; ++j) {
      const int k = k0 + j;                  // k = tap*C2 + ci
      const int tap = k / C2_, ci = k % C2_;
      Bf[ent * 2 + j] = w3[((ct * 16 + n) * C2_ + ci) * 9 + tap];
    }
  }
  if (tid < C3_) bLDS[tid] = b3[tid];
  __syncthreads();

  const int lane = tid & 31;
  const int wv = tid >> 5;
  const int ct = wv & 1;                      // out-channel tile
  const int wrow = wv >> 1;                   // row within block
  const int n = lane & 15;
  const int shiftbase = (lane >> 4) << 3;
  const int wpix = blockIdx.x * 16 + n;
  const int h = blockIdx.y * 2 + wrow;        // may be 25 (inactive)
  const int b = blockIdx.z;
  const float bias = bLDS[ct * 16 + n];

  v8f mem = {0.f,0.f,0.f,0.f,0.f,0.f,0.f,0.f};
  v8f sum = {0.f,0.f,0.f,0.f,0.f,0.f,0.f,0.f};

  for (int t = 0; t < T_; ++t) {
    unsigned wdw[18];
#pragma unroll
    for (int dy = -1; dy <= 1; ++dy)
#pragma unroll
      for (int dx = -1; dx <= 1; ++dx) {
        const int tap = (dy + 1) * 3 + (dx + 1);
        const int hh = h + dy, ww = wpix + dx;
        unsigned v0 = 0u, v1 = 0u;
        if (hh >= 0 && hh < H3 && ww >= 0 && ww < W3) {
          const uint2 u =
              *(const uint2*)(P2 + (((size_t)(t * B_ + b) * H3 + hh) * W3 + ww) * 2);
          v0 = u.x; v1 = u.y;
        }
        wdw[tap * 2 + 0] = v0;
        wdw[tap * 2 + 1] = v1;
      }
    v8f acc = {bias,bias,bias,bias,bias,bias,bias,bias};
#pragma unroll
    for (int kk = 0; kk < NK; ++kk) {
      const unsigned wsel = wdw[(kk >> 2) * 2 + ((kk & 3) >> 1)];
      const int s = ((kk & 1) << 4) + shiftbase;
      v2f a;
      a.x = (float)((wsel >> s) & 0xFu) * 0.25f;
      a.y = (float)((wsel >> (s + 4)) & 0xFu) * 0.25f;
      const v2f bb =
          *reinterpret_cast<const v2f*>(&Bf[((ct * NK + kk) * 32 + lane) * 2]);
      acc = __builtin_amdgcn_wmma_f32_16x16x4_f32(false, a, false, bb,
                                                  (short)0, acc, false, false);
    }
#pragma unroll
    for (int r = 0; r < 8; ++r) {
      const float mm = BETA * mem[r] + acc[r];
      const bool sp = (mm >= THR);
      sum[r] += sp ? 1.f : 0.f;
      mem[r] = mm - (sp ? THR : 0.f);
      const unsigned bal = (unsigned)__ballot(sp);
      if (lane == 0) {
        maskLDS[wrow][ct][r] = bal & 0xFFFFu;
        maskLDS[wrow][ct][8 + r] = bal >> 16;
      }
    }
    __syncthreads();
    if (tid < 8) {
      const int pw = blockIdx.x * 8 + tid;
      const int ph = blockIdx.y;
      if (pw < W4 && ph < H4) {
        const int px = 2 * tid;
        unsigned mr[2][2][2];  // [row][dx][cotile]
#pragma unroll
        for (int rr = 0; rr < 2; ++rr)
#pragma unroll
          for (int dxx = 0; dxx < 2; ++dxx)
#pragma unroll
            for (int cc = 0; cc < 2; ++cc)
              mr[rr][dxx][cc] = maskLDS[rr][cc][px + dxx];
        unsigned wordv[4] = {0u, 0u, 0u, 0u};
#pragma unroll
        for (int c = 0; c < C3_; ++c) {
          const int hf = c >> 4, bit = c & 15;
          const unsigned cnt =
              ((mr[0][0][hf] >> bit) & 1u) + ((mr[0][1][hf] >> bit) & 1u) +
              ((mr[1][0][hf] >> bit) & 1u) + ((mr[1][1][hf] >> bit) & 1u);
          wordv[c >> 3] |= cnt << (4 * (c & 7));
        }
        const size_t base = (((size_t)(t * B_ + b) * H4 + ph) * W4 + pw) * 4;
#pragma unroll
        for (int i = 0; i < 4; ++i) P3[base + i] = wordv[i];
      }
    }
    __syncthreads();
  }
  if (h < H3) {
#pragma unroll
    for (int r = 0; r < 8; ++r) {
      const int m = r + ((lane >> 4) << 3);
      const int w = blockIdx.x * 16 + m;
      if (w < W3)
        S3[(size_t)(b * C3_ + ct * 16 + n) * (H3 * W3) + h * W3 + w] = sum[r];
    }
  }
}

// =====================================================================
// Layer 4 (WMMA): conv(32->32) + LIF, sums only. K = 288 -> 72 steps.
// Block = 2 waves = 2 out-channel tiles over the same 16 pixels.
// =====================================================================
__global__ __launch_bounds__(64) void k_layer4(const unsigned* __restrict__ P3,
    const float* __restrict__ w4, const float* __restrict__ b4,
    float* __restrict__ S4) {
  constexpr int NK = 288 / 4;  // 72
  __shared__ float Bf[2 * NK * 32 * 2];  // 9216 floats (36.9 KB)
  __shared__ float bLDS[C4_];
  const int tid = threadIdx.x;
  for (int ent = tid; ent < 2 * NK * 32; ent += 64) {
    const int el = ent & 31;
    const int kk = (ent >> 5) % NK;
    const int ct = (ent >> 5) / NK;
    const int n = el & 15;
    const int k0 = kk * 4 + ((el >> 4) << 1);
#pragma unroll
    for (int j = 0; j < 2; ++j) {
      const int k = k0 + j;                  // k = tap*C3 + ci
      const int tap = k / C3_, ci = k % C3_;
      Bf[ent * 2 + j] = w4[((ct * 16 + n) * C3_ + ci) * 9 + tap];
    }
  }
  if (tid < C4_) bLDS[tid] = b4[tid];
  __syncthreads();

  const int lane = tid & 31;
  const int ct = tid >> 5;
  const int n = lane & 15;
  const int shiftbase = (lane >> 4) << 3;
  const int p = blockIdx.x * 16 + n;          // flat pixel over 540
  const int b = blockIdx.y;
  const bool act = p < H4 * W4;
  const int h = p / W4, w = p % W4;
  const float bias = bLDS[ct * 16 + n];

  v8f mem = {0.f,0.f,0.f,0.f,0.f,0.f,0.f,0.f};
  v8f sum = {0.f,0.f,0.f,0.f,0.f,0.f,0.f,0.f};

  for (int t = 0; t < T_; ++t) {
    unsigned wdw[36];
#pragma unroll
    for (int dy = -1; dy <= 1; ++dy)
#pragma unroll
      for (int dx = -1; dx <= 1; ++dx) {
        const int tap = (dy + 1) * 3 + (dx + 1);
        const int hh = h + dy, ww = w + dx;
        uint4 u = {0u, 0u, 0u, 0u};
        if (act && hh >= 0 && hh < H4 && ww >= 0 && ww < W4)
          u = *(const uint4*)(P3 + (((size_t)(t * B_ + b) * H4 + hh) * W4 + ww) * 4);
        wdw[tap * 4 + 0] = u.x;
        wdw[tap * 4 + 1] = u.y;
        wdw[tap * 4 + 2] = u.z;
        wdw[tap * 4 + 3] = u.w;
      }
    v8f acc = {bias,bias,bias,bias,bias,bias,bias,bias};
#pragma unroll
    for (int kk = 0; kk < NK; ++kk) {
      const unsigned wsel = wdw[(kk >> 3) * 4 + ((kk & 7) >> 1)];
      const int s = ((kk & 1) << 4) + shiftbase;
      v2f a;
      a.x = (float)((wsel >> s) & 0xFu) * 0.25f;
      a.y = (float)((wsel >> (s + 4)) & 0xFu) * 0.25f;
      const v2f bb =
          *reinterpret_cast<const v2f*>(&Bf[((ct * NK + kk) * 32 + lane) * 2]);
      acc = __builtin_amdgcn_wmma_f32_16x16x4_f32(false, a, false, bb,
                                                  (short)0, acc, false, false);
    }
#pragma unroll
    for (int r = 0; r < 8; ++r) {
      const float mm = BETA * mem[r] + acc[r];
      const bool sp = (mm >= THR);
      sum[r] += sp ? 1.f : 0.f;
      mem[r] = mm - (sp ? THR : 0.f);
    }
  }
#pragma unroll
  for (int r = 0; r < 8; ++r) {
    const int m = r + ((lane >> 4) << 3);
    const int pp = blockIdx.x * 16 + m;
    if (pp < H4 * W4)
      S4[(size_t)(b * C4_ + ct * 16 + n) * (H4 * W4) + pp] = sum[r];
  }
}

// =====================================================================
// Decoder: bilinear-upsampled concat(88 ch) -> dense(32)+relu -> dense(1)
// GEMM (M=73400, K=88, N=32) via V_WMMA_F32_16X16X4_F32.
// =====================================================================
DEV float bilin(const float* __restrict__ S, int C, int HW, int Wl,
                int b, int c, int y0, int y1, int x0, int x1, float ty, float tx) {
  const float* Sp = S + (size_t)(b * C + c) * HW;
  const float v00 = Sp[y0 * Wl + x0], v01 = Sp[y0 * Wl + x1];
  const float v10 = Sp[y1 * Wl + x0], v11 = Sp[y1 * Wl + x1];
  return (1.f - ty) * ((1.f - tx) * v00 + tx * v01) +
         ty * ((1.f - tx) * v10 + tx * v11);
}

DEV void iparams(int o, int insz, int outsz, int& i0, int& i1, float& t) {
  const float f = (o + 0.5f) * (float)insz / (float)outsz - 0.5f;
  const int i = (int)floorf(f);
  t = f - (float)i;
  i0 = min(max(i, 0), insz - 1);
  i1 = min(max(i + 1, 0), insz - 1);
}

__global__ __launch_bounds__(32) void k_decoder(const float* __restrict__ S1,
    const float* __restrict__ S2, const float* __restrict__ S3,
    const float* __restrict__ S4,
    const float* __restrict__ dw1, const float* __restrict__ db1,
    const float* __restrict__ dw2, const float* __restrict__ db2,
    float* __restrict__ out) {
  constexpr int NP = B_ * H1 * W1;  // 73400
  const int lane = threadIdx.x;
  const int tile = blockIdx.x;
  const int mrow = lane & 15;
  int p = tile * 16 + mrow;
  if (p >= NP) p = NP - 1;
  const int b = p / (H1 * W1);
  const int rem = p - b * (H1 * W1);
  const int h = rem / W1;
  const int w = rem - h * W1;

  int y0a, y1a, x0a, x1a; float tya, txa;
  int y0b, y1b, x0b, x1b; float tyb, txb;
  int y0c, y1c, x0c, x1c; float tyc, txc;
  iparams(h, H2, H1, y0a, y1a, tya); iparams(w, W2, W1, x0a, x1a, txa);
  iparams(h, H3, H1, y0b, y1b, tyb); iparams(w, W3, W1, x0b, x1b, txb);
  iparams(h, H4, H1, y0c, y1c, tyc); iparams(w, W4, W1, x0c, x1c, txc);

  const int khalf = (lane >> 4) * 2;
  v8f acc0 = {0.f, 0.f, 0.f, 0.f, 0.f, 0.f, 0.f, 0.f};
  v8f acc1 = {0.f, 0.f, 0.f, 0.f, 0.f, 0.f, 0.f, 0.f};

  for (int kk = 0; kk < CC / 4; ++kk) {
    const int kbase = kk * 4 + khalf;
    v2f a, bb0, bb1;
#pragma unroll
    for (int j = 0; j < 2; ++j) {
      const int k = kbase + j;
      float val;
      if (k < 8) {
        val = S1[(size_t)(b * C1_ + k) * (H1 * W1) + h * W1 + w];
      } else if (k < 24) {
        val = bilin(S2, C2_, H2 * W2, W2, b, k - 8, y0a, y1a, x0a, x1a, tya, txa);
      } else if (k < 56) {
        val = bilin(S3, C3_, H3 * W3, W3, b, k - 24, y0b, y1b, x0b, x1b, tyb, txb);
      } else {
        val = bilin(S4, C4_, H4 * W4, W4, b, k - 56, y0c, y1c, x0c, x1c, tyc, txc);
      }
      a[j] = val;
      bb0[j] = dw1[(size_t)mrow * CC + k];
      bb1[j] = dw1[(size_t)(16 + mrow) * CC + k];
    }
    acc0 = __builtin_amdgcn_wmma_f32_16x16x4_f32(false, a, false, bb0,
                                                 (short)0, acc0, false, false);
    acc1 = __builtin_amdgcn_wmma_f32_16x16x4_f32(false, a, false, bb1,
                                                 (short)0, acc1, false, false);
  }

  const float bias0 = db1[mrow], bias1 = db1[16 + mrow];
  const float d20 = dw2[mrow],  d21 = dw2[16 + mrow];
  const float d2b = db2[0];
#pragma unroll
  for (int r = 0; r < 8; ++r) {
    const float h0 = fmaxf(acc0[r] + bias0, 0.f);
    const float h1 = fmaxf(acc1[r] + bias1, 0.f);
    float v = fmaf(h0, d20, h1 * d21);
    v += __shfl_xor(v, 1, 32);
    v += __shfl_xor(v, 2, 32);
    v += __shfl_xor(v, 4, 32);
    v += __shfl_xor(v, 8, 32);
    const int m = (lane < 16) ? r : (8 + r);
    const int pp = tile * 16 + m;
    if (mrow == 0 && pp < NP) out[pp] = v + d2b;
  }
}

// =====================================================================
extern "C" void kernel_launch(void* const* d_in, const int* in_sizes, int n_in,
                              void* d_out, int out_size, void* d_ws, size_t ws_size,
                              hipStream_t stream) {
  (void)in_sizes; (void)n_in; (void)out_size; (void)ws_size;
  const float* x   = (const float*)d_in[0];
  const float* w1  = (const float*)d_in[1];
  const float* b1  = (const float*)d_in[2];
  const float* w2  = (const float*)d_in[3];
  const float* b2  = (const float*)d_in[4];
  const float* w3  = (const float*)d_in[5];
  const float* b3  = (const float*)d_in[6];
  const float* w4  = (const float*)d_in[7];
  const float* b4  = (const float*)d_in[8];
  const float* dw1 = (const float*)d_in[9];
  const float* db1 = (const float*)d_in[10];
  const float* dw2 = (const float*)d_in[11];
  const float* db2 = (const float*)d_in[12];

  char* ws = (char*)d_ws;
  unsigned* P1 = (unsigned*)(ws + oP1);
  unsigned* P2 = (unsigned*)(ws + oP2);
  unsigned* P3 = (unsigned*)(ws + oP3);
  float* S1 = (float*)(ws + oS1);
  float* S2 = (float*)(ws + oS2);
  float* S3 = (float*)(ws + oS3);
  float* S4 = (float*)(ws + oS4);
  float* out = (float*)d_out;

  dim3 g1((W1 + 15) / 16, (H1 + 7) / 8, B_);   // 23 x 13 x 2
  k_layer1<<<g1, dim3(128), 0, stream>>>(x, w1, b1, P1, S1);

  dim3 g2((W2 + 15) / 16, H2 / 2, B_);         // 12 x 25 x 2
  k_layer2<<<g2, dim3(64), 0, stream>>>(P1, w2, b2, P2, S2);

  dim3 g3((W3 + 15) / 16, (H3 + 1) / 2, B_);   // 6 x 13 x 2
  k_layer3<<<g3, dim3(128), 0, stream>>>(P2, w3, b3, P3, S3);

  dim3 g4((H4 * W4 + 15) / 16, B_, 1);         // 34 x 2
  k_layer4<<<g4, dim3(64), 0, stream>>>(P3, w4, b4, S4);

  const int NP = B_ * H1 * W1;
  k_decoder<<<dim3((NP + 15) / 16), dim3(32), 0, stream>>>(
      S1, S2, S3, S4, dw1, db1, dw2, db2, out);
}